// LBD_48765058679468
// MI455X (gfx1250) — compile-verified
//
#include <hip/hip_runtime.h>
#include <hip/hip_bf16.h>

typedef __attribute__((ext_vector_type(2))) float v2f;
typedef __attribute__((ext_vector_type(8))) float v8f;

#define NBATCH   131072
#define NHID     64
#define NBINS    10
#define FEATSTR  65              // NUM_HIDDEN + 1
#define NQ       64
#define EPSV     1e-06f
#define LOG2E    1.44269504088896340736f

// Raw hardware transcendentals: v_exp_f32 (2^x) and v_log_f32 (log2 x).
__device__ __forceinline__ float fast_exp2(float x) { return __builtin_amdgcn_exp2f(x); }
__device__ __forceinline__ float fast_log2(float x) { return __builtin_amdgcn_logf(x); }

// Block: 256 threads = 8 waves (wave32). Each wave computes two 16-sample
// WMMA tiles (diag(U.V^T), diag(U.U^T), diag(V.V^T) in full fp32 via
// V_WMMA_F32_16X16X4_F32 chains), then each thread handles one sample's
// beta-CDF quadrature.
__global__ __launch_bounds__(256) void lbd_beta_mass_kernel(
    const int* __restrict__ uid, const int* __restrict__ iid,
    const float* __restrict__ uf, const float* __restrict__ vf,
    const float* __restrict__ ubin, const float* __restrict__ ibin,
    const float* __restrict__ ualpha, const float* __restrict__ ubeta,
    const float* __restrict__ ialpha, const float* __restrict__ ibeta,
    const float* __restrict__ gbias,
    float* __restrict__ out, int n)
{
    __shared__ float sdot[256];
    __shared__ float suu[256];
    __shared__ float svv[256];

    const int tid  = threadIdx.x;
    const int lane = tid & 31;
    const int wid  = tid >> 5;
    const int blockBase = blockIdx.x * 256;

    // ---------------- Phase 1: WMMA dot/norm tiles ----------------
    const int r = lane & 15;      // tile row this lane feeds
    const int h = lane >> 4;      // K half (cols 2h, 2h+1 of each K=4 chunk)

    #pragma unroll
    for (int tile = 0; tile < 2; ++tile) {
        const int sBase = blockBase + wid * 32 + tile * 16;
        const int su = (sBase + r < n) ? sBase + r : 0;
        const long urow = (long)uid[su] * FEATSTR;
        const long irow = (long)iid[su] * FEATSTR;
        const float* Ur = uf + urow;
        const float* Vr = vf + irow;

        v8f cuv = {}; v8f cuu = {}; v8f cvv = {};
        #pragma unroll
        for (int k = 0; k < 16; ++k) {
            const int c = 4 * k + 2 * h;
            v2f aU; aU.x = Ur[c]; aU.y = Ur[c + 1];
            v2f aV; aV.x = Vr[c]; aV.y = Vr[c + 1];
            // Same per-lane data is valid as operand A (U rows) and B (V^T).
            cuv = __builtin_amdgcn_wmma_f32_16x16x4_f32(false, aU, false, aV,
                                                        (short)0, cuv, false, false);
            cuu = __builtin_amdgcn_wmma_f32_16x16x4_f32(false, aU, false, aU,
                                                        (short)0, cuu, false, false);
            cvv = __builtin_amdgcn_wmma_f32_16x16x4_f32(false, aV, false, aV,
                                                        (short)0, cvv, false, false);
        }

        // Extract diagonal of the 16x16 fp32 accumulators:
        // element (m,m): m<8 -> VGPR m, lane m ; m>=8 -> VGPR m-8, lane m+16.
        const int slot = wid * 32 + tile * 16;
        #pragma unroll
        for (int j = 0; j < 8; ++j) {
            if (lane == j) {
                sdot[slot + j] = cuv[j]; suu[slot + j] = cuu[j]; svv[slot + j] = cvv[j];
            }
            if (lane == j + 24) {
                sdot[slot + j + 8] = cuv[j]; suu[slot + j + 8] = cuu[j]; svv[slot + j + 8] = cvv[j];
            }
        }
    }
    __syncthreads();

    // ---------------- Phase 2: per-sample beta-CDF mass ----------------
    const int s = blockBase + tid;
    if (s >= n) return;

    const float dot = sdot[tid];
    const float uu  = suu[tid];
    const float vv  = svv[tid];
    const int u  = uid[s];
    const int it = iid[s];

    const float lenp = __builtin_sqrtf(uu * vv);
    float mu = 0.5f + 0.5f * dot / fmaxf(lenp, EPSV);
    mu = fminf(fmaxf(mu, EPSV), 1.0f - EPSV);
    const float ups = fmaxf(fabsf(dot), EPSV);
    float alpha = fmaxf(0.01f, mu * ups);
    float beta  = fmaxf(0.01f, ups - alpha);
    const float gb = gbias[0];
    alpha = fmaxf(0.01f, gb + alpha + ualpha[u] + ialpha[it]);
    beta  = fmaxf(0.01f, gb + beta  + ubeta[u]  + ibeta[it]);
    const float am1 = alpha - 1.0f;
    const float bm1 = beta  - 1.0f;

    // Softmax over bins -> interior CDF edges x[0..8]
    float terms[NBINS];
    float tsum = 0.0f;
    #pragma unroll
    for (int j = 0; j < NBINS; ++j) {
        terms[j] = fast_exp2((ubin[u * NBINS + j] + ibin[it * NBINS + j]) * LOG2E);
        tsum += terms[j];
    }
    const float inv_ts = 1.0f / tsum;
    float x[NBINS - 1], lx[NBINS - 1];
    float acc = 0.0f;
    #pragma unroll
    for (int j = 0; j < NBINS - 1; ++j) {
        acc += terms[j];
        x[j]  = acc * inv_ts;
        lx[j] = fast_log2(x[j]);
    }

    // Trapezoid sums. dt cancels in Bx/B; endpoint terms are exactly zero
    // (alpha,beta > 1 and the t<=x mask kills t=1), so interior sums suffice.
    float Bab = 0.0f;
    float Bx[NBINS - 1];
    #pragma unroll
    for (int j = 0; j < NBINS - 1; ++j) Bx[j] = 0.0f;

    const float invQ = 1.0f / (float)(NQ - 1);
    for (int i = 1; i <= NQ - 2; ++i) {
        const float t  = (float)i * invQ;
        const float lt  = fast_log2(t);
        const float l1t = fast_log2(1.0f - t);
        Bab += fast_exp2(am1 * lt) * fast_exp2(bm1 * l1t);
        #pragma unroll
        for (int j = 0; j < NBINS - 1; ++j) {
            if (t <= x[j]) {
                const float tx = t * x[j];
                Bx[j] += fast_exp2(am1 * (lt + lx[j])) *
                         fast_exp2(bm1 * fast_log2(1.0f - tx));
            }
        }
    }

    const float invB = 1.0f / Bab;
    float* o = out + (size_t)s * NBINS;
    float prev = 0.0f;
    #pragma unroll
    for (int j = 0; j < NBINS - 1; ++j) {
        const float c = Bx[j] * invB;
        o[j] = c - prev;
        prev = c;
    }
    o[NBINS - 1] = 1.0f - prev;
}

extern "C" void kernel_launch(void* const* d_in, const int* in_sizes, int n_in,
                              void* d_out, int out_size, void* d_ws, size_t ws_size,
                              hipStream_t stream) {
    const int*   uid    = (const int*)d_in[0];
    const int*   iid    = (const int*)d_in[1];
    const float* uf     = (const float*)d_in[2];
    const float* vf     = (const float*)d_in[3];
    const float* ubin   = (const float*)d_in[4];
    const float* ibin   = (const float*)d_in[5];
    const float* ualpha = (const float*)d_in[6];
    const float* ubeta  = (const float*)d_in[7];
    const float* ialpha = (const float*)d_in[8];
    const float* ibeta  = (const float*)d_in[9];
    const float* gbias  = (const float*)d_in[10];
    float* out = (float*)d_out;

    const int n = in_sizes[0];            // BATCH
    const int blocks = (n + 255) / 256;   // 512 for BATCH=131072
    lbd_beta_mass_kernel<<<blocks, 256, 0, stream>>>(
        uid, iid, uf, vf, ubin, ibin, ualpha, ubeta, ialpha, ibeta, gbias,
        out, n);
}